// GCN_15659450761582
// MI455X (gfx1250) — compile-verified
//
#include <hip/hip_runtime.h>
#include <math.h>

// GCN 2-layer inference for MI455X (gfx1250, wave32).
// GEMMs use native fp32 WMMA V_WMMA_F32_16X16X4_F32 (16x16 tile per wave).
// Edge aggregation uses global_atomic_add_f32; working set fits the 192MB L2.

#define NN   50000     // nodes
#define NE   800000    // edges
#define KIN  512       // in feats
#define HF   256       // hidden feats
#define NC   40        // classes
#define NCP  48        // classes padded to multiple of 16

typedef __attribute__((ext_vector_type(2))) float v2f;
typedef __attribute__((ext_vector_type(8))) float v8f;

// ---------------- utility kernels ----------------

__global__ void zero_f32(float* __restrict__ p, long n) {
  long i = (long)blockIdx.x * blockDim.x + threadIdx.x;
  long stride = (long)gridDim.x * blockDim.x;
  for (; i < n; i += stride) p[i] = 0.0f;
}

__global__ void degree_k(const int* __restrict__ src, const int* __restrict__ dst,
                         float* __restrict__ degO, float* __restrict__ degI) {
  int i = blockIdx.x * blockDim.x + threadIdx.x;
  int stride = gridDim.x * blockDim.x;
  for (; i < NE; i += stride) {
    unsafeAtomicAdd(&degO[src[i]], 1.0f);
    unsafeAtomicAdd(&degI[dst[i]], 1.0f);
  }
}

__global__ void invsqrt_k(float* __restrict__ p, int n) {
  int i = blockIdx.x * blockDim.x + threadIdx.x;
  if (i < n) {
    float v = p[i];
    p[i] = (v > 0.0f) ? rsqrtf(v) : 0.0f;
  }
}

// ---------------- WMMA GEMM 1: [NN,KIN] @ [KIN,HF] -> [NN,HF], row-scaled ----------------
// One wave computes one 16x16 output tile; K stepped by 4 via v_wmma_f32_16x16x4_f32.
// A frag (16x4 f32): lane half h holds K = {2h, 2h+1} in v[0],v[1] -> contiguous float2.
// B frag (4x16 f32): lane l holds column n0+l, rows kb, kb+1 (kb = kk+2h).
// D: VGPR j -> row m0 + j + 8h, col n0 + l.

__global__ void __launch_bounds__(128)
gemm1_k(const float* __restrict__ X, const float* __restrict__ W,
        const float* __restrict__ invO, float* __restrict__ Out) {
  const int K = KIN, N = HF;
  int tile = blockIdx.x * 4 + (threadIdx.x >> 5);   // 3125*16 = 50000 tiles
  int m0 = (tile >> 4) << 4;                        // 16 n-tiles per row
  int n0 = (tile & 15) << 4;
  int lane = threadIdx.x & 31;
  int half = lane >> 4;
  int l    = lane & 15;

  const float* arow = X + (size_t)(m0 + l) * K + 2 * half;
  const float* bcol = W + n0 + l;

  v8f acc = {};
#pragma unroll 4
  for (int kk = 0; kk < K; kk += 4) {
    v2f a = *(const v2f*)(arow + kk);               // 8B-aligned float2
    int kb = kk + 2 * half;
    v2f b;
    b.x = bcol[(size_t)kb * N];
    b.y = bcol[(size_t)(kb + 1) * N];
    acc = __builtin_amdgcn_wmma_f32_16x16x4_f32(
        false, a, false, b, (short)0, acc, false, false);
  }

#pragma unroll
  for (int j = 0; j < 8; ++j) {
    int row = m0 + j + 8 * half;
    Out[(size_t)row * N + n0 + l] = acc[j] * invO[row];   // fold D_out^{-1/2}
  }
}

// ---------------- WMMA GEMM 2: [NN,HF] @ [HF,NC] -> [NN,NCP], row-scaled ----------------

__global__ void __launch_bounds__(128)
gemm2_k(const float* __restrict__ Hin, const float* __restrict__ W,
        const float* __restrict__ invO, float* __restrict__ Out) {
  const int K = HF;
  const int TILES = (NN / 16) * (NCP / 16);         // 3125 * 3 = 9375
  int tile = blockIdx.x * 4 + (threadIdx.x >> 5);
  if (tile >= TILES) return;                        // wave-uniform exit (EXEC stays all-1s)
  int m0 = (tile / 3) << 4;
  int n0 = (tile % 3) << 4;
  int lane = threadIdx.x & 31;
  int half = lane >> 4;
  int l    = lane & 15;

  const float* arow = Hin + (size_t)(m0 + l) * K + 2 * half;
  int col = n0 + l;
  bool cok = (col < NC);

  v8f acc = {};
#pragma unroll 4
  for (int kk = 0; kk < K; kk += 4) {
    v2f a = *(const v2f*)(arow + kk);
    int kb = kk + 2 * half;
    v2f b;
    b.x = cok ? W[(size_t)kb * NC + col] : 0.0f;    // predicate padded columns
    b.y = cok ? W[(size_t)(kb + 1) * NC + col] : 0.0f;
    acc = __builtin_amdgcn_wmma_f32_16x16x4_f32(
        false, a, false, b, (short)0, acc, false, false);
  }

#pragma unroll
  for (int j = 0; j < 8; ++j) {
    int row = m0 + j + 8 * half;
    Out[(size_t)row * NCP + n0 + l] = acc[j] * invO[row];
  }
}

// ---------------- edge scatter (segment_sum over dst) ----------------
// Layer 1: one block per edge, 256 coalesced feature lanes.

__global__ void scatter1_k(const int* __restrict__ src, const int* __restrict__ dst,
                           const float* __restrict__ H, float* __restrict__ AGG) {
  int e = blockIdx.x;
  int f = threadIdx.x;
  int s = src[e], d = dst[e];
  unsafeAtomicAdd(&AGG[(size_t)d * HF + f], H[(size_t)s * HF + f]);
}

// Layer 2: grid-stride over (edge, class).

__global__ void scatter2_k(const int* __restrict__ src, const int* __restrict__ dst,
                           const float* __restrict__ H2, float* __restrict__ AGG2) {
  long i = (long)blockIdx.x * blockDim.x + threadIdx.x;
  long total = (long)NE * NC;
  long stride = (long)gridDim.x * blockDim.x;
  for (; i < total; i += stride) {
    int e = (int)(i / NC);
    int f = (int)(i - (long)e * NC);
    int s = src[e], d = dst[e];
    unsafeAtomicAdd(&AGG2[(size_t)d * NCP + f], H2[(size_t)s * NCP + f]);
  }
}

// ---------------- epilogues ----------------

__global__ void epi1_k(float* __restrict__ AGG, const float* __restrict__ invI,
                       const float* __restrict__ b1) {
  long i = (long)blockIdx.x * blockDim.x + threadIdx.x;
  long total = (long)NN * HF;
  long stride = (long)gridDim.x * blockDim.x;
  for (; i < total; i += stride) {
    int row = (int)(i >> 8);          // HF == 256
    int col = (int)(i & 255);
    float v = AGG[i] * invI[row] + b1[col];
    AGG[i] = fmaxf(v, 0.0f);          // ReLU
  }
}

__global__ void final_k(const float* __restrict__ AGG2, const float* __restrict__ invI,
                        const float* __restrict__ b2, float* __restrict__ out) {
  int r = blockIdx.x * blockDim.x + threadIdx.x;
  if (r >= NN) return;
  float inv = invI[r];
  float v[NC];
  float m = -1e30f;
#pragma unroll
  for (int c = 0; c < NC; ++c) {
    float t = fmaxf(AGG2[(size_t)r * NCP + c] * inv + b2[c], 0.0f);
    v[c] = t;
    m = fmaxf(m, t);
  }
  float s = 0.0f;
#pragma unroll
  for (int c = 0; c < NC; ++c) s += __expf(v[c] - m);
  float ls = __logf(s);
#pragma unroll
  for (int c = 0; c < NC; ++c) out[(size_t)r * NC + c] = v[c] - m - ls;
}

// ---------------- driver ----------------

extern "C" void kernel_launch(void* const* d_in, const int* in_sizes, int n_in,
                              void* d_out, int out_size, void* d_ws, size_t ws_size,
                              hipStream_t stream) {
  const float* x  = (const float*)d_in[0];
  const int* esrc = (const int*)d_in[1];
  const int* edst = (const int*)d_in[2];
  const float* W1 = (const float*)d_in[3];
  const float* b1 = (const float*)d_in[4];
  const float* W2 = (const float*)d_in[5];
  const float* b2 = (const float*)d_in[6];
  float* out = (float*)d_out;
  float* ws  = (float*)d_ws;

  // workspace layout (floats): [degO|degI|H1|AGG1|H2]; AGG2 overlays H1 (dead by then)
  float* degO = ws;                                 // NN  -> becomes invO
  float* degI = ws + NN;                            // NN  -> becomes invI
  float* H1   = ws + 2 * (size_t)NN;                // NN*HF
  float* AGG1 = H1 + (size_t)NN * HF;               // NN*HF
  float* H2   = AGG1 + (size_t)NN * HF;             // NN*NCP
  float* AGG2 = H1;                                 // NN*NCP (reuse)

  // degrees -> D^{-1/2}
  zero_f32<<<256, 256, 0, stream>>>(degO, 2L * NN);
  degree_k<<<1024, 256, 0, stream>>>(esrc, edst, degO, degI);
  invsqrt_k<<<(2 * NN + 255) / 256, 256, 0, stream>>>(ws, 2 * NN);

  // layer 1
  gemm1_k<<<(NN / 16) * (HF / 16) / 4, 128, 0, stream>>>(x, W1, degO, H1);
  zero_f32<<<2048, 256, 0, stream>>>(AGG1, (long)NN * HF);
  scatter1_k<<<NE, HF, 0, stream>>>(esrc, edst, H1, AGG1);
  epi1_k<<<2048, 256, 0, stream>>>(AGG1, degI, b1);

  // layer 2
  gemm2_k<<<((NN / 16) * (NCP / 16) + 3) / 4, 128, 0, stream>>>(AGG1, W2, degO, H2);
  zero_f32<<<512, 256, 0, stream>>>(AGG2, (long)NN * NCP);
  scatter2_k<<<4096, 256, 0, stream>>>(esrc, edst, H2, AGG2);

  // bias + ReLU + log_softmax
  final_k<<<(NN + 255) / 256, 256, 0, stream>>>(AGG2, degI, b2, out);
}